// VolumeSDFRenderer_87179246174517
// MI455X (gfx1250) — compile-verified
//
#include <hip/hip_runtime.h>

#ifndef __has_builtin
#define __has_builtin(x) 0
#endif

#if __has_builtin(__builtin_amdgcn_global_load_async_to_lds_b128)
#define USE_ASYNC_LDS 1
#else
#define USE_ASYNC_LDS 0
#endif

#define P_SAMPLES 128
#define WAVES 16
#define TPB (WAVES * 32)

#if USE_ASYNC_LDS
// Match the builtin's prototype exactly: (v4i AS1*, v4i AS3*, imm, imm)
typedef int v4i __attribute__((ext_vector_type(4)));
typedef __attribute__((address_space(1))) v4i gv4i;  // global
typedef __attribute__((address_space(3))) v4i lv4i;  // LDS
#endif

__device__ __forceinline__ void wait_async_all() {
#if __has_builtin(__builtin_amdgcn_s_wait_asynccnt)
  __builtin_amdgcn_s_wait_asynccnt(0);
#else
  asm volatile("s_wait_asynccnt 0x0" ::: "memory");
#endif
}

// One wave32 per ray; lane l owns samples 4l..4l+3.
__global__ __launch_bounds__(TPB) void VolumeSDFRenderer_kernel(
    const float* __restrict__ sdf,     // [R, 128]
    const float* __restrict__ color,   // [R, 128, 3]
    const float* __restrict__ depth,   // [R, 128]
    float* __restrict__ out,           // [R, 3]
    int R)
{
  __shared__ __align__(16) float lcolor[WAVES * P_SAMPLES * 3];  // 24 KB

  const int lane = threadIdx.x & 31;
  const int wave = threadIdx.x >> 5;
  const int ray  = blockIdx.x * WAVES + wave;
  if (ray >= R) return;

  const size_t rbase = (size_t)ray * P_SAMPLES;
  const float* gcol  = color + rbase * 3;          // 1536 B per ray
  float*       wcol  = &lcolor[wave * (P_SAMPLES * 3)];
  (void)wcol;

#if USE_ASYNC_LDS
  // Prefetch this ray's color stream into LDS: 96 chunks x 16 B, 3 per lane.
  // Tracked on ASYNCcnt; overlaps with the transcendental/scan math below.
#pragma unroll
  for (int k = 0; k < 3; ++k) {
    const int c = lane + 32 * k;  // 16-byte chunk index
    __builtin_amdgcn_global_load_async_to_lds_b128(
        (gv4i*)(gcol + 4 * c),
        (lv4i*)(wcol + 4 * c),
        0, 0);
  }
#endif

  // --- coalesced 16B loads of sdf & depth -------------------------------
  const float4 s4 = *(const float4*)(sdf + rbase + 4 * lane);
  const float4 z4 = *(const float4*)(depth + rbase + 4 * lane);

  // density = 50 * sigmoid(e) * sigmoid(-e),  e = -50*sdf   (overflow-safe)
  float dens[4];
  {
    const float sv[4] = {s4.x, s4.y, s4.z, s4.w};
#pragma unroll
    for (int j = 0; j < 4; ++j) {
      const float e  = -50.0f * sv[j];
      const float sg = 1.0f / (1.0f + __expf(-e));  // sigmoid(e)
      dens[j] = 50.0f * sg * (1.0f - sg);
    }
  }

  // deltas: forward differences; neighbor's first depth via shuffle.
  const float zn   = __shfl_down(z4.x, 1);
  const float del0 = z4.y - z4.x;
  const float del1 = z4.z - z4.y;
  const float del2 = z4.w - z4.z;
  const float del3 = (lane == 31) ? 1e10f : (zn - z4.w);

  // tau and in-lane inclusive prefix of 4 samples
  const float t0 = dens[0] * del0;
  const float t1 = dens[1] * del1;
  const float t2 = dens[2] * del2;
  const float t3 = dens[3] * del3;
  const float p0 = t0, p1 = p0 + t1, p2 = p1 + t2, p3 = p2 + t3;

  // wave32 inclusive scan of per-lane totals -> exclusive lane offset
  float incl = p3;
#pragma unroll
  for (int off = 1; off < 32; off <<= 1) {
    const float n = __shfl_up(incl, off);
    if (lane >= off) incl += n;
  }
  const float excl = incl - p3;

  // weights via telescoping: w_j = exp(-excl_j) - exp(-excl_{j+1})
  const float E0 = __expf(-excl);
  const float E1 = __expf(-(excl + p0));
  const float E2 = __expf(-(excl + p1));
  const float E3 = __expf(-(excl + p2));
  const float E4 = __expf(-incl);
  const float w0 = E0 - E1, w1 = E1 - E2, w2 = E2 - E3, w3 = E3 - E4;

  // --- consume color (from LDS if async path is live) -------------------
  float4 c0, c1, c2;
#if USE_ASYNC_LDS
  wait_async_all();
  c0 = *(const float4*)(wcol + 12 * lane + 0);
  c1 = *(const float4*)(wcol + 12 * lane + 4);
  c2 = *(const float4*)(wcol + 12 * lane + 8);
#else
  c0 = *(const float4*)(gcol + 12 * lane + 0);
  c1 = *(const float4*)(gcol + 12 * lane + 4);
  c2 = *(const float4*)(gcol + 12 * lane + 8);
#endif

  // samples 4l..4l+3 rgb are interleaved across the three float4s
  float pr = w0 * c0.x + w1 * c0.w + w2 * c1.z + w3 * c2.y;
  float pg = w0 * c0.y + w1 * c1.x + w2 * c1.w + w3 * c2.z;
  float pb = w0 * c0.z + w1 * c1.y + w2 * c2.x + w3 * c2.w;

  // wave32 reduction of the 3 channels
#pragma unroll
  for (int off = 16; off > 0; off >>= 1) {
    pr += __shfl_xor(pr, off);
    pg += __shfl_xor(pg, off);
    pb += __shfl_xor(pb, off);
  }

  if (lane == 0) {
    float* o = out + (size_t)ray * 3;
    o[0] = pr;
    o[1] = pg;
    o[2] = pb;
  }
}

extern "C" void kernel_launch(void* const* d_in, const int* in_sizes, int n_in,
                              void* d_out, int out_size, void* d_ws, size_t ws_size,
                              hipStream_t stream) {
  const float* sdf   = (const float*)d_in[0];
  const float* color = (const float*)d_in[1];
  const float* depth = (const float*)d_in[2];
  float* out = (float*)d_out;

  const int R      = in_sizes[0] / P_SAMPLES;      // 65536
  const int blocks = (R + WAVES - 1) / WAVES;      // 4096 blocks x 512 threads

  VolumeSDFRenderer_kernel<<<blocks, TPB, 0, stream>>>(sdf, color, depth, out, R);
}